// ComplexProjection_62861141344417
// MI455X (gfx1250) — compile-verified
//
#include <hip/hip_runtime.h>
#include <math.h>

// Problem constants (match reference)
#define B_TOT 32768
#define R_TOT 16
#define S_TOT 128
#define P_TOT 128

typedef float v2f __attribute__((ext_vector_type(2)));
typedef float v4f __attribute__((ext_vector_type(4)));
typedef float v8f __attribute__((ext_vector_type(8)));

// LDS layout for W: k-quad interleaved so a WMMA B-fragment is one b64 load.
//   f(k, p) = (k>>2)*512 + p*4 + (k&3)      (floats)
// Lane needs (W[k0+2*kh][col], W[k0+2*kh+1][col]) -> contiguous pair at
//   (k0>>2)*512 + col*4 + 2*kh.
// Banks: lanes 0-15 -> {4L,4L+1}, lanes 16-31 -> {4(L-16)+2, 4(L-16)+3}
// => all 64 banks, zero conflicts. Total 128*128*4 = 64 KB.
__global__ __launch_bounds__(256)
void ComplexProjection_62861141344417_kernel(const float* __restrict__ xr,
                                             const float* __restrict__ xi,
                                             const float* __restrict__ w,
                                             float* __restrict__ out)
{
    __shared__ float wLds[S_TOT * P_TOT]; // 64 KB

    const int r    = blockIdx.y;
    const int tid  = threadIdx.x;
    const int wave = tid >> 5;
    const int lane = tid & 31;

    // ---- Stage W[r] into LDS in k-quad interleaved layout ----
    {
        const v4f* __restrict__ src = (const v4f*)(w + (size_t)r * (S_TOT * P_TOT));
        #pragma unroll
        for (int i = 0; i < (S_TOT * P_TOT / 4) / 256; ++i) {
            const int gi = i * 256 + tid;     // v4f index (coalesced 16B loads)
            const v4f d  = src[gi];
            const int k  = gi >> 5;           // source row 0..127
            const int p  = (gi & 31) * 4;     // source col base
            float* q = wLds + (k >> 2) * 512 + (k & 3) + p * 4;
            q[0]  = d.x;
            q[4]  = d.y;
            q[8]  = d.z;
            q[12] = d.w;
        }
    }
    __syncthreads();

    const int b0    = blockIdx.x * 128 + wave * 16; // first M row of this wave's strip
    const int mrow  = lane & 15;                    // M (for A) / N (for B,C,D)
    const int khalf = lane >> 4;                    // 0: K{0,1}, 1: K{2,3}

    // Per-lane LDS base for B fragments
    const float* __restrict__ bptr = wLds + (mrow * 4 + 2 * khalf);

    // A-fragment row base: x[b0+mrow, r, :], contiguous in s
    const float* __restrict__ arow_r = xr + ((size_t)(b0 + mrow) * R_TOT + r) * S_TOT;
    const float* __restrict__ arow_i = xi + ((size_t)(b0 + mrow) * R_TOT + r) * S_TOT;

    v8f acc_re[8], acc_im[8];
    const v8f zacc = {0.f, 0.f, 0.f, 0.f, 0.f, 0.f, 0.f, 0.f};
    #pragma unroll
    for (int nt = 0; nt < 8; ++nt) { acc_re[nt] = zacc; acc_im[nt] = zacc; }

    #pragma unroll 2
    for (int k0 = 0; k0 < S_TOT; k0 += 4) {
        // A frags (16x4 f32): lane holds A[mrow, k0+2*khalf .. +1] -> one b64 load
        const v2f a_re = *(const v2f*)(arow_r + k0 + 2 * khalf);
        const v2f a_im = *(const v2f*)(arow_i + k0 + 2 * khalf);

        #pragma unroll
        for (int nt = 0; nt < 8; ++nt) {
            // B frag: single contiguous b64 from LDS, offset is an immediate
            const v2f bfrag = *(const v2f*)(bptr + (k0 >> 2) * 512 + nt * 64);

            acc_re[nt] = __builtin_amdgcn_wmma_f32_16x16x4_f32(
                false, a_re, false, bfrag, (short)0, acc_re[nt], false, false);
            acc_im[nt] = __builtin_amdgcn_wmma_f32_16x16x4_f32(
                false, a_im, false, bfrag, (short)0, acc_im[nt], false, false);
        }
    }

    // ---- |z| epilogue + store. C/D layout: N = lane&15, M = vgpr + 8*(lane>=16) ----
    float* __restrict__ obase = out + ((size_t)b0 * R_TOT + r) * P_TOT;
    #pragma unroll
    for (int nt = 0; nt < 8; ++nt) {
        const int p = nt * 16 + mrow;
        #pragma unroll
        for (int i = 0; i < 8; ++i) {
            const int m  = khalf * 8 + i;
            const float re = acc_re[nt][i];
            const float im = acc_im[nt][i];
            obase[(size_t)m * (R_TOT * P_TOT) + p] = sqrtf(re * re + im * im);
        }
    }
}

extern "C" void kernel_launch(void* const* d_in, const int* in_sizes, int n_in,
                              void* d_out, int out_size, void* d_ws, size_t ws_size,
                              hipStream_t stream) {
    const float* xr = (const float*)d_in[0];  // x_real  (B, R, S)
    const float* xi = (const float*)d_in[1];  // x_imag  (B, R, S)
    const float* w  = (const float*)d_in[2];  // projection (R, S, P)
    float* out = (float*)d_out;               // (B, R, P) float32

    dim3 grid(B_TOT / 128, R_TOT, 1);
    dim3 block(256, 1, 1);
    ComplexProjection_62861141344417_kernel<<<grid, block, 0, stream>>>(xr, xi, w, out);
}